// SpanRepresentationLayer_41506563948815
// MI455X (gfx1250) — compile-verified
//
#include <hip/hip_runtime.h>
#include <hip/hip_bf16.h>

// ---------------------------------------------------------------------------
// SpanRepresentationLayer, MI455X-optimized.
//
// Reference does: cat = [h[start], h[end]] (B,N,512); out = cat @ W^T + b.
// We factor the GEMM through the concat:
//   P1[b,s,q] = sum_k hid[b,s,k] * W[q, k]      (q = t*H+h, W row stride 512)
//   P2[b,s,q] = sum_k hid[b,s,k] * W[q, 256+k]
//   out[b,n,q] = P1[b,start(n),q] + P2[b,end(n),q] + bias[q]
// GEMM shrinks from 102 GFLOP -> 8.6 GFLOP; P1/P2 (64MB) live in the 192MB L2
// so the 12x-reuse gather phase is bounded only by the 400MB output stream
// (~17us at 23.3 TB/s).
// ---------------------------------------------------------------------------

typedef __attribute__((ext_vector_type(2))) float v2f;
typedef __attribute__((ext_vector_type(4))) float v4f;
typedef __attribute__((ext_vector_type(8))) float v8f;

#define BATCH   8
#define SEQ     1024
#define HID     256
#define NTYPES  4
#define MSPAN   12
#define NSPANS  12222      // sum_s min(12, 1024-s)
#define NFULL   12156      // (SEQ-MSPAN+1)*MSPAN
#define GM      (BATCH*SEQ)     // 8192 rows
#define QD      (NTYPES*HID)    // 1024 cols per half (j=0: W[:, :256], j=1: W[:, 256:])
#define GK      HID             // 256

// ---------------------------------------------------------------------------
// Kernel 1: P[j][m][q] = sum_k A[m,k] * W[q*512 + j*256 + k]
// One wave computes a 16x64 strip: 4 accumulators, A-fragment reused 4x.
// fp32 WMMA fragment layout (ISA 7.12.2, 32-bit 16x4):
//   lane holds row/col = lane%16, K = 2*(lane/16) + {0,1}  -> float2 loads.
// ---------------------------------------------------------------------------
__global__ __launch_bounds__(256)
void span_gemm_kernel(const float* __restrict__ A,   // hidden: 8192 x 256
                      const float* __restrict__ W,   // 1024 rows x 512
                      float* __restrict__ P)         // [2][8192][1024]
{
    const int lane  = threadIdx.x & 31;
    const int wave  = threadIdx.x >> 5;
    const int widx  = blockIdx.x * 8 + wave;   // 16384 waves total
    const int m_tile  = widx & 511;            // 512 tiles of 16 rows
    const int n_strip = widx >> 9;             // 32 strips of 64 cols

    const int m0 = m_tile * 16;
    const int c0 = n_strip * 64;
    const int j  = c0 / QD;                    // which W half (64 | 1024 -> constant per strip)
    const int qb = c0 % QD;

    const int lmod = lane & 15;
    const int lhi  = lane >> 4;                // 0 or 1 -> K offset 0 or 2

    const float* a_ptr = A + (size_t)(m0 + lmod) * GK + 2 * lhi;
    const float* b_ptr0 = W + (size_t)(qb + 0  + lmod) * (2 * GK) + j * GK + 2 * lhi;
    const float* b_ptr1 = W + (size_t)(qb + 16 + lmod) * (2 * GK) + j * GK + 2 * lhi;
    const float* b_ptr2 = W + (size_t)(qb + 32 + lmod) * (2 * GK) + j * GK + 2 * lhi;
    const float* b_ptr3 = W + (size_t)(qb + 48 + lmod) * (2 * GK) + j * GK + 2 * lhi;

    v8f acc0 = {}, acc1 = {}, acc2 = {}, acc3 = {};

#pragma unroll 4
    for (int k = 0; k < GK; k += 4) {
        v2f a  = *(const v2f*)(a_ptr  + k);
        v2f b0 = *(const v2f*)(b_ptr0 + k);
        v2f b1 = *(const v2f*)(b_ptr1 + k);
        v2f b2 = *(const v2f*)(b_ptr2 + k);
        v2f b3 = *(const v2f*)(b_ptr3 + k);
        acc0 = __builtin_amdgcn_wmma_f32_16x16x4_f32(false, a, false, b0, (short)0, acc0, false, false);
        acc1 = __builtin_amdgcn_wmma_f32_16x16x4_f32(false, a, false, b1, (short)0, acc1, false, false);
        acc2 = __builtin_amdgcn_wmma_f32_16x16x4_f32(false, a, false, b2, (short)0, acc2, false, false);
        acc3 = __builtin_amdgcn_wmma_f32_16x16x4_f32(false, a, false, b3, (short)0, acc3, false, false);
    }

    // C/D layout: VGPR r, lanes 0-15 -> M=r, N=lane; lanes 16-31 -> M=r+8.
    float* pbase = P + (size_t)j * GM * QD;
#pragma unroll
    for (int r = 0; r < 8; ++r) {
        const int m = m0 + r + 8 * lhi;
        float* row = pbase + (size_t)m * QD + qb + lmod;
        row[0]  = acc0[r];
        row[16] = acc1[r];
        row[32] = acc2[r];
        row[48] = acc3[r];
    }
}

// ---------------------------------------------------------------------------
// Kernel 2: out[b, n*T+t, h] = P1[b,s(n),t*H+h] + P2[b,e(n),t*H+h] + bias[t*H+h]
// One block per (b, n); 256 threads x float4 = 1024 contiguous outputs.
// P1/P2 reads hit L2 (12x reuse); output written non-temporally.
// ---------------------------------------------------------------------------
__global__ __launch_bounds__(256)
void span_gather_kernel(const float* __restrict__ P,
                        const float* __restrict__ bias,
                        float* __restrict__ out)
{
    const int n = blockIdx.x;
    const int b = blockIdx.y;

    int s, e;
    if (n < NFULL) {
        s = n / MSPAN;
        e = s + (n % MSPAN);
    } else {
        int r = n - NFULL;
        s = SEQ - MSPAN + 1;
        int cnt = MSPAN - 1;
        while (r >= cnt) { r -= cnt; --cnt; ++s; }
        e = s + r;
    }

    const v4f* p1 = (const v4f*)(P + (size_t)(b * SEQ + s) * QD);
    const v4f* p2 = (const v4f*)(P + (size_t)GM * QD + (size_t)(b * SEQ + e) * QD);
    const v4f* bi = (const v4f*)bias;
    v4f*       o  = (v4f*)(out + (size_t)(b * NSPANS + n) * QD);

    const int u = threadIdx.x;          // 256 float4 lanes covering 1024 floats
    v4f r = p1[u] + p2[u] + bi[u];
    __builtin_nontemporal_store(r, &o[u]);
}

extern "C" void kernel_launch(void* const* d_in, const int* in_sizes, int n_in,
                              void* d_out, int out_size, void* d_ws, size_t ws_size,
                              hipStream_t stream) {
    (void)in_sizes; (void)n_in; (void)out_size; (void)ws_size;
    const float* hidden = (const float*)d_in[0];   // (8,1024,256)
    const float* W      = (const float*)d_in[1];   // (4,256,512) -> rows q=t*H+h, len 512
    const float* bias   = (const float*)d_in[2];   // (4,256)
    float* P   = (float*)d_ws;                     // [2][8192][1024] fp32 = 64 MB
    float* out = (float*)d_out;                    // (8, 48888, 256)

    // 16384 waves / 8 per block
    span_gemm_kernel<<<2048, 256, 0, stream>>>(hidden, W, P);

    dim3 g2(NSPANS, BATCH);
    span_gather_kernel<<<g2, 256, 0, stream>>>(P, bias, out);
}